// GraphConv_936302871047
// MI455X (gfx1250) — compile-verified
//
#include <hip/hip_runtime.h>

// ---------------------------------------------------------------------------
// GraphConv for MI455X (gfx1250):
//   phase 1: scatter-add   neigh[dst] += x[src]          (wave-per-edge, f32 atomics)
//   phase 2: fused GEMM    out = neigh*Wn^T + x*Wr^T + (bn+br)   (V_WMMA_F32_16X16X4_F32)
// ---------------------------------------------------------------------------

typedef __attribute__((ext_vector_type(2))) float v2f;
typedef __attribute__((ext_vector_type(8))) float v8f;

#define C_IN 128   // feature width (K)
#define C_OUT 128  // output width (N)

// -------------------- phase 1: edge scatter-add ----------------------------
// One wave (32 lanes) per edge; each lane moves 4 consecutive floats.
__global__ void __launch_bounds__(256) graphconv_scatter(
    const float* __restrict__ x,
    const long long* __restrict__ edge_index,  // [2, E] int64
    float* __restrict__ neigh,
    int num_edges)
{
    const long long gid = (long long)blockIdx.x * 256 + threadIdx.x;
    const int e    = (int)(gid >> 5);
    const int lane = (int)(gid & 31);
    if (e >= num_edges) return;

    const int s = (int)edge_index[e];
    const int d = (int)edge_index[(long long)num_edges + e];

    const float4 v = *(const float4*)&x[(long long)s * C_IN + lane * 4];
    float* dst = &neigh[(long long)d * C_IN + lane * 4];

    __hip_atomic_fetch_add(&dst[0], v.x, __ATOMIC_RELAXED, __HIP_MEMORY_SCOPE_AGENT);
    __hip_atomic_fetch_add(&dst[1], v.y, __ATOMIC_RELAXED, __HIP_MEMORY_SCOPE_AGENT);
    __hip_atomic_fetch_add(&dst[2], v.z, __ATOMIC_RELAXED, __HIP_MEMORY_SCOPE_AGENT);
    __hip_atomic_fetch_add(&dst[3], v.w, __ATOMIC_RELAXED, __HIP_MEMORY_SCOPE_AGENT);
}

// -------------------- phase 2: fused dual GEMM + bias ----------------------
// Block: 256 threads = 8 waves. blockIdx.y selects a 16-wide N slice;
// wave w of blockIdx.x handles M-tile (blockIdx.x*8 + w), i.e. 16 rows.
// Weight slices for the block's 16 output columns are staged in LDS.
__global__ void __launch_bounds__(256) graphconv_gemm(
    const float* __restrict__ x,
    const float* __restrict__ neigh,
    const float* __restrict__ W_neigh,  // [C_OUT, C_IN] row-major (nn.Linear)
    const float* __restrict__ W_root,
    const float* __restrict__ b_neigh,
    const float* __restrict__ b_root,
    float* __restrict__ out,
    int num_nodes)
{
    __shared__ __align__(16) float sW[2][16][C_IN];  // [matrix][n in slice][k]
    __shared__ float sBias[16];

    const int tid = threadIdx.x;
    const int nb  = blockIdx.y * 16;  // base output column of this block

    // Stage 16-column slices of both weight matrices: 2*16*128 floats = 16 KB.
    for (int i = tid; i < 1024; i += 256) {       // 1024 float4 loads
        const int t  = i >> 9;                    // which matrix
        const int r  = i & 511;
        const int n  = r >> 5;                    // row within slice (0..15)
        const int k4 = (r & 31) << 2;             // k offset (multiple of 4)
        const float* src = (t == 0 ? W_neigh : W_root) + (size_t)(nb + n) * C_IN + k4;
        *(float4*)&sW[t][n][k4] = *(const float4*)src;
    }
    if (tid < 16) sBias[tid] = b_neigh[nb + tid] + b_root[nb + tid];
    __syncthreads();

    const int lane  = tid & 31;
    const int wave  = tid >> 5;
    const int mbase = (blockIdx.x * 8 + wave) * 16;
    if (mbase >= num_nodes) return;               // wave-uniform: EXEC stays all-ones

    // A-fragment addressing (32-bit A 16x4 layout, ISA 7.12.2):
    //   lanes 0-15 : row = lane,    holds K = kb+0, kb+1
    //   lanes 16-31: row = lane-16, holds K = kb+2, kb+3
    const int arow = mbase + (lane & 15);
    const int kh   = (lane >> 4) << 1;            // 0 or 2
    const int ncol = lane & 15;

    const float* An = neigh + (size_t)arow * C_IN;
    const float* Ax = x     + (size_t)arow * C_IN;

    v8f acc = {};
    #pragma unroll 4
    for (int kb = 0; kb < C_IN; kb += 4) {
        const v2f aN = *(const v2f*)&An[kb + kh];
        const v2f aX = *(const v2f*)&Ax[kb + kh];
        // B 4x16 fragment from LDS: B[k][n] = W[n][k] (Linear does x @ W^T)
        const v2f bN = *(const v2f*)&sW[0][ncol][kb + kh];
        const v2f bR = *(const v2f*)&sW[1][ncol][kb + kh];

        acc = __builtin_amdgcn_wmma_f32_16x16x4_f32(
                  false, aN, false, bN, (short)0, acc, false, false);
        acc = __builtin_amdgcn_wmma_f32_16x16x4_f32(
                  false, aX, false, bR, (short)0, acc, false, false);
    }

    // C/D layout: VGPR j -> row mbase + j (+8 for upper half-lanes), col = nb + (lane&15)
    const int moff = (lane >> 4) << 3;            // 0 or 8
    #pragma unroll
    for (int j = 0; j < 8; ++j) {
        const int row = mbase + moff + j;
        out[(size_t)row * C_OUT + nb + ncol] = acc[j] + sBias[ncol];
    }
}

// ---------------------------------------------------------------------------
extern "C" void kernel_launch(void* const* d_in, const int* in_sizes, int n_in,
                              void* d_out, int out_size, void* d_ws, size_t ws_size,
                              hipStream_t stream) {
    const float*     x  = (const float*)d_in[0];
    const long long* ei = (const long long*)d_in[1];
    const float*     Wn = (const float*)d_in[2];
    const float*     bn = (const float*)d_in[3];
    const float*     Wr = (const float*)d_in[4];
    const float*     br = (const float*)d_in[5];
    float* out   = (float*)d_out;
    float* neigh = (float*)d_ws;   // [num_nodes, 128] f32 scratch

    const int num_nodes = in_sizes[0] / C_IN;
    const int num_edges = in_sizes[1] / 2;

    // neigh = 0
    hipMemsetAsync(neigh, 0, (size_t)num_nodes * C_IN * sizeof(float), stream);

    // phase 1: scatter-add, one wave per edge
    {
        const long long total = (long long)num_edges * 32;
        const int blocks = (int)((total + 255) / 256);
        graphconv_scatter<<<blocks, 256, 0, stream>>>(x, ei, neigh, num_edges);
    }

    // phase 2: fused dual GEMM + bias
    {
        const int mtiles = (num_nodes + 15) / 16;          // 6250
        dim3 grid((mtiles + 7) / 8, C_OUT / 16);           // (782, 8)
        graphconv_gemm<<<grid, 256, 0, stream>>>(x, neigh, Wn, Wr, bn, br, out, num_nodes);
    }
}